// LinearLUT_40097814675910
// MI455X (gfx1250) — compile-verified
//
#include <hip/hip_runtime.h>

// ---- Problem constants (from reference) ----
#define BATCH   256
#define IN_F    64
#define OUT_F   64
#define KBITS   6
#define KK      64                 // 2^KBITS basis size
#define TABLES  (IN_F * OUT_F)     // 4096
#define TPO     (TABLES / OUT_F)   // 64 tables summed into each output
#define XSTRIDE 65                 // padded LDS row stride: bank-conflict-free

typedef __attribute__((ext_vector_type(2))) float v2f;
typedef __attribute__((ext_vector_type(8))) float v8f;

// Dynamic LDS layout:
//   xp : BATCH x XSTRIDE floats (padded copy of x)    66560 B
//   wl : TPO x KK floats (this block's weights)       16384 B
//   ml : TPO x KBITS ints (this block's gather idx)    1536 B
#define XP_BYTES  (BATCH * XSTRIDE * 4)
#define WL_BYTES  (TPO * KK * 4)
#define ML_BYTES  (TPO * KBITS * 4)
#define SMEM_BYTES (XP_BYTES + WL_BYTES + ML_BYTES)

__global__ __launch_bounds__(256) void linearlut_kernel(
    const float* __restrict__ x,
    const int*   __restrict__ mask,
    const float* __restrict__ weight,
    const float* __restrict__ bias,
    float*       __restrict__ out)
{
  extern __shared__ char smem[];
  float* xp = (float*)smem;
  float* wl = (float*)(smem + XP_BYTES);
  int*   ml = (int*)(smem + XP_BYTES + WL_BYTES);

  const int o   = blockIdx.x;      // output feature owned by this block
  const int tid = threadIdx.x;

  // Warm L2 for this block's weight slab / masks (global_prefetch_b8 on gfx1250)
  __builtin_prefetch(weight + (size_t)o * TPO * KK, 0, 1);
  __builtin_prefetch(mask   + (size_t)o * TPO * KBITS, 0, 1);

  // ---- Stage x: coalesced global reads; stride-65 rows -> conflict-free LDS
  #pragma unroll 4
  for (int it = 0; it < (BATCH * IN_F) / 256; ++it) {
    int idx = it * 256 + tid;
    int b = idx >> 6, f = idx & (IN_F - 1);
    xp[b * XSTRIDE + f] = x[idx];
  }
  // ---- Stage this block's 64 tables of weights (16 KB)
  #pragma unroll 4
  for (int it = 0; it < (TPO * KK) / 256; ++it) {
    int idx = it * 256 + tid;
    wl[idx] = weight[o * (TPO * KK) + idx];
  }
  // ---- Stage this block's gather indices (384 ints)
  ml[tid] = mask[o * (TPO * KBITS) + tid];
  if (tid < TPO * KBITS - 256)
    ml[256 + tid] = mask[o * (TPO * KBITS) + 256 + tid];
  __syncthreads();

  const float bias_o = bias[o];

  const int lane = tid & 31;
  const int wv   = tid >> 5;           // wave id 0..7
  const int bl   = lane & 15;          // batch-within-tile (A-matrix row M)
  const int rq   = (lane < 16) ? 0 : 2;// A-matrix K slots: lanes<16 -> K0,K1; else K2,K3
  const v2f onesB = {1.0f, 1.0f};      // B = all-ones 4x16 (layout independent)

  // Each wave: 16 batches; two tiles per wave cover all 256 batches.
  for (int tile = wv; tile < 16; tile += 8) {
    const int bb = tile * 16 + bl;     // this lane's batch
    v8f c = {0.f, 0.f, 0.f, 0.f, 0.f, 0.f, 0.f, 0.f};

    // 16 steps x 4 tables = 64 tables; WMMA C-tile accumulates the r-reduction.
    for (int r0 = 0; r0 < TPO; r0 += 4) {
      v2f a;
      #pragma unroll
      for (int q = 0; q < 2; ++q) {
        const int r = r0 + rq + q;     // table within this output group

        // Gather 6 features for (bb, table r): broadcast pairs, distinct banks
        float xv[6];
        #pragma unroll
        for (int i = 0; i < 6; ++i)
          xv[i] = xp[bb * XSTRIDE + ml[r * KBITS + i]];

        float fn[6], fp[6];
        #pragma unroll
        for (int i = 0; i < 6; ++i) {
          fn[i] = 0.5f - 0.5f * xv[i];
          fp[i] = 0.5f + 0.5f * xv[i];
        }

        // P_lo over x0..x2 (bit0->x0), P_hi over x3..x5 (bit0->x3)
        float plo[8], phi[8];
        {
          float b00 = fn[0] * fn[1], b01 = fp[0] * fn[1];
          float b10 = fn[0] * fp[1], b11 = fp[0] * fp[1];
          plo[0] = b00 * fn[2]; plo[1] = b01 * fn[2];
          plo[2] = b10 * fn[2]; plo[3] = b11 * fn[2];
          plo[4] = b00 * fp[2]; plo[5] = b01 * fp[2];
          plo[6] = b10 * fp[2]; plo[7] = b11 * fp[2];
        }
        {
          float b00 = fn[3] * fn[4], b01 = fp[3] * fn[4];
          float b10 = fn[3] * fp[4], b11 = fp[3] * fp[4];
          phi[0] = b00 * fn[5]; phi[1] = b01 * fn[5];
          phi[2] = b10 * fn[5]; phi[3] = b11 * fn[5];
          phi[4] = b00 * fp[5]; phi[5] = b01 * fp[5];
          phi[6] = b10 * fp[5]; phi[7] = b11 * fp[5];
        }

        // out_t = sum_hi P_hi[hi] * (sum_lo w[8*hi+lo] * P_lo[lo])
        float acc = 0.0f;
        #pragma unroll
        for (int hi = 0; hi < 8; ++hi) {
          float s = 0.0f;
          #pragma unroll
          for (int lo = 0; lo < 8; ++lo)
            s += wl[r * KK + hi * 8 + lo] * plo[lo];
          acc += s * phi[hi];
        }
        a[q] = acc;
      }
      // D[m,n] = C[m,n] + sum_k A[m,k]  (B = ones) : accumulates the r-sum
      c = __builtin_amdgcn_wmma_f32_16x16x4_f32(
              /*neg_a=*/false, a, /*neg_b=*/false, onesB,
              /*c_mod=*/(short)0, c, /*reuse_a=*/false, /*reuse_b=*/false);
    }

    // C/D layout: lanes 0-15 VGPR j = row M=j; lanes 16-31 VGPR j = row M=j+8.
    // All columns are identical (B = ones), so lane 0 / lane 16 hold the sums.
    if (lane == 0) {
      #pragma unroll
      for (int j = 0; j < 8; ++j)
        out[(tile * 16 + j) * OUT_F + o] = c[j] + bias_o;
    } else if (lane == 16) {
      #pragma unroll
      for (int j = 0; j < 8; ++j)
        out[(tile * 16 + 8 + j) * OUT_F + o] = c[j] + bias_o;
    }
  }
}

extern "C" void kernel_launch(void* const* d_in, const int* in_sizes, int n_in,
                              void* d_out, int out_size, void* d_ws, size_t ws_size,
                              hipStream_t stream) {
  (void)in_sizes; (void)n_in; (void)d_ws; (void)ws_size; (void)out_size;
  const float* x    = (const float*)d_in[0];
  const int*   mask = (const int*)  d_in[1];
  const float* w    = (const float*)d_in[2];
  const float* bias = (const float*)d_in[3];
  float* out = (float*)d_out;

  linearlut_kernel<<<dim3(OUT_F), dim3(256), SMEM_BYTES, stream>>>(
      x, mask, w, bias, out);
}